// HashSideOut_1322849927726
// MI455X (gfx1250) — compile-verified
//
#include <hip/hip_runtime.h>
#include <cmath>

typedef __attribute__((ext_vector_type(2))) float v2f;
typedef __attribute__((ext_vector_type(8))) float v8f;

#define TABLE_NUM 16
#define TABLE_SIZE 65536
#define NPTS 65536           // 256*256 points per batch
#define BATCH 8
#define PI2 2654435761u

struct Scales { float s[TABLE_NUM]; };

// ---------------------------------------------------------------------------
// Kernel 1: style = s @ affine_w^T + affine_b ; demod = rsqrt(sum((w*style)^2)+eps)
// writes wd[b][o][i] = weight[o][i] * style[b][i] * demod[b][o]   (8*3*32 floats)
// ---------------------------------------------------------------------------
__global__ __launch_bounds__(256) void style_kernel(
    const float* __restrict__ s, const float* __restrict__ aw,
    const float* __restrict__ ab, const float* __restrict__ weight,
    float* __restrict__ wd)
{
    __shared__ float st[BATCH][32];
    __shared__ float dm[BATCH][3];
    const int tid = threadIdx.x;
    const int b = tid >> 5;          // 0..7
    const int i = tid & 31;          // 0..31
    float acc = ab[i];
    const float* sr = s + (size_t)b * 512;
    const float* ar = aw + (size_t)i * 512;
    for (int k = 0; k < 512; ++k) acc += sr[k] * ar[k];
    st[b][i] = acc;
    __syncthreads();
    if (tid < 24) {
        const int bb = tid / 3, o = tid % 3;
        float sum = 0.f;
        for (int k = 0; k < 32; ++k) {
            float v = weight[o * 32 + k] * st[bb][k];
            sum += v * v;
        }
        dm[bb][o] = 1.0f / sqrtf(sum + 1e-8f);
    }
    __syncthreads();
    for (int j = tid; j < BATCH * 3 * 32; j += 256) {
        const int bb = j / 96, r = j % 96, o = r / 32, k = r % 32;
        wd[j] = weight[o * 32 + k] * st[bb][k] * dm[bb][o];
    }
}

// ---------------------------------------------------------------------------
// Kernel 2: hash-grid gather + bilinear -> feats[32] per point (1 pt/lane),
// then modulated projection to 3 channels via chained f32 WMMA 16x16x4.
// Block = 256 threads = 8 waves; each wave owns 32 points; grid = 2048 blocks.
// ---------------------------------------------------------------------------
__global__ __launch_bounds__(256) void hashside_main(
    const float* __restrict__ x, const float* __restrict__ coords,
    const float* __restrict__ wd, const float* __restrict__ bias,
    float* __restrict__ out, Scales sc)
{
    __shared__ float lfeat[8][32 * 33];      // per-wave 32 pts x 32 feats, stride 33
    const int tid  = threadIdx.x;
    const int wave = tid >> 5;
    const int lane = tid & 31;
    const int gtid = blockIdx.x * 256 + tid;
    const int b    = gtid >> 16;             // / 65536 (block never straddles b)
    const int p    = gtid & (NPTS - 1);

    const float2 c = ((const float2*)coords)[(size_t)b * NPTS + p];
    const float2* __restrict__ tb =
        (const float2*)x + (size_t)b * TABLE_NUM * TABLE_SIZE;

    float feats[32];
#pragma unroll 4
    for (int l = 0; l < TABLE_NUM; ++l) {
        const float sl = sc.s[l];
        const float px = c.x * sl, py = c.y * sl;
        const float fx = floorf(px), fy = floorf(py);
        const float wx = px - fx, wy = py - fy;
        const unsigned ix = (unsigned)(int)fx;
        const unsigned iy = (unsigned)(int)fy;
        const float2* __restrict__ t2 = tb + (size_t)l * TABLE_SIZE;
        const unsigned hy0 = iy * PI2;
        const unsigned hy1 = (iy + 1u) * PI2;
        const unsigned i00 = (ix        ^ hy0) & (TABLE_SIZE - 1u);
        const unsigned i10 = ((ix + 1u) ^ hy0) & (TABLE_SIZE - 1u);
        const unsigned i01 = (ix        ^ hy1) & (TABLE_SIZE - 1u);
        const unsigned i11 = ((ix + 1u) ^ hy1) & (TABLE_SIZE - 1u);
        const float2 g00 = t2[i00];
        const float2 g10 = t2[i10];
        const float2 g01 = t2[i01];
        const float2 g11 = t2[i11];
        const float w00 = (1.f - wx) * (1.f - wy);
        const float w10 = wx * (1.f - wy);
        const float w01 = (1.f - wx) * wy;
        const float w11 = wx * wy;
        feats[2 * l]     = w00 * g00.x + w10 * g10.x + w01 * g01.x + w11 * g11.x;
        feats[2 * l + 1] = w00 * g00.y + w10 * g10.y + w01 * g01.y + w11 * g11.y;
    }

    // Stash per-point features for fragment rebuild (stride 33 -> conflict-free).
    {
        float* fr = &lfeat[wave][lane * 33];
#pragma unroll
        for (int f = 0; f < 32; ++f) fr[f] = feats[f];
    }
    __syncthreads();

    // B fragment (4x16 per K-chunk): column N = out channel (3 valid), rows K.
    // Layout mirrors the documented 16x4 A layout: VGPR0 = K0 (lanes 0-15) /
    // K2 (lanes 16-31); VGPR1 = K1 / K3.
    const int n     = lane & 15;
    const int khalf = (lane >> 4) << 1;          // 0 or 2
    const float* wdr = wd + b * 96;              // wd[b][o][i], o-major rows of 32
    v2f bf[8];
#pragma unroll
    for (int kk = 0; kk < 8; ++kk) {
        const int k0 = 4 * kk + khalf;
        bf[kk].x = (n < 3) ? wdr[n * 32 + k0]     : 0.f;
        bf[kk].y = (n < 3) ? wdr[n * 32 + k0 + 1] : 0.f;
    }
    const float bn = (n < 3) ? bias[n] : 0.f;
    const int m = lane & 15;

#pragma unroll
    for (int t = 0; t < 2; ++t) {                // two 16-point tiles per wave
        const float* ab = &lfeat[wave][(t * 16 + m) * 33 + khalf];
        v8f acc = {0.f, 0.f, 0.f, 0.f, 0.f, 0.f, 0.f, 0.f};
#pragma unroll
        for (int kk = 0; kk < 8; ++kk) {         // K = 32 as 8 chained K=4 steps
            v2f a;
            a.x = ab[4 * kk];
            a.y = ab[4 * kk + 1];
            acc = __builtin_amdgcn_wmma_f32_16x16x4_f32(
                false, a, false, bf[kk], (short)0, acc, false, false);
        }
        if (n < 3) {
            const int pbase = ((blockIdx.x * 256) & (NPTS - 1)) + wave * 32 + t * 16;
#pragma unroll
            for (int r = 0; r < 8; ++r) {
                const int mm = (lane < 16) ? r : (r + 8);
                out[((size_t)(b * 3 + n)) * NPTS + pbase + mm] = acc[r] + bn;
            }
        }
    }
}

// ---------------------------------------------------------------------------
extern "C" void kernel_launch(void* const* d_in, const int* in_sizes, int n_in,
                              void* d_out, int out_size, void* d_ws, size_t ws_size,
                              hipStream_t stream) {
    const float* x       = (const float*)d_in[0];   // [8,16,131072]
    const float* coords  = (const float*)d_in[1];   // [8,65536,2]
    const float* s       = (const float*)d_in[2];   // [8,512]
    const float* aw      = (const float*)d_in[3];   // [32,512]
    const float* ab      = (const float*)d_in[4];   // [32]
    const float* weight  = (const float*)d_in[5];   // [3,32]
    const float* bias    = (const float*)d_in[6];   // [3]
    float* out = (float*)d_out;                     // [8,3,256,256]
    float* wd  = (float*)d_ws;                      // 768 floats scratch

    Scales sc;
    const double g = exp((log(256.0) - log(16.0)) / 15.0);
    for (int l = 0; l < TABLE_NUM; ++l)
        sc.s[l] = (float)floor(16.0 * pow(g, (double)l));

    style_kernel<<<1, 256, 0, stream>>>(s, aw, ab, weight, wd);
    hashside_main<<<(BATCH * NPTS) / 256, 256, 0, stream>>>(x, coords, wd, bias, out, sc);
}